// SoftDotAttention_28982439313604
// MI455X (gfx1250) — compile-verified
//
#include <hip/hip_runtime.h>
#include <stdint.h>

// Problem constants (from reference): B=64, L=4096, D=256
#define B_ 64
#define L_ 4096
#define D_ 256
#define SEG 32            // segments per batch row
#define SEG_ROWS (L_ / SEG)   // 128 context rows per block

typedef __attribute__((ext_vector_type(2))) float    f32x2;
typedef __attribute__((ext_vector_type(8))) float    v8f;
typedef __attribute__((ext_vector_type(4))) unsigned u32x4;
typedef __attribute__((ext_vector_type(8))) int      i32x8;
typedef __attribute__((ext_vector_type(4))) int      i32x4;

// ---------------------------------------------------------------------------
// Kernel 1: target = tanh(input @ W_in^T)   [64,256] = [64,256] x [256,256]^T
// One wave computes one 16x16 output tile with V_WMMA_F32_16X16X4_F32.
// A layout (16x4 f32): lane = m + 16*kh ; v0 = A[m][2kh], v1 = A[m][2kh+1]
// B layout (4x16 f32): lane = n + 16*kh ; v0 = B[2kh][n], v1 = B[2kh+1][n]
// C/D layout: vgpr r, lane: row = r + 8*(lane>>4), col = lane&15
// ---------------------------------------------------------------------------
__global__ __launch_bounds__(32) void sda_qt_kernel(
    const float* __restrict__ input, const float* __restrict__ W_in,
    float* __restrict__ target) {
  const int m0 = blockIdx.x * 16;        // over B
  const int n0 = blockIdx.y * 16;        // over D
  const int lane = threadIdx.x;
  const int mr = lane & 15;              // row (A) / col (B) index
  const int kh = lane >> 4;              // 0 or 1

  v8f c = {};
  const float* arow = input + (size_t)(m0 + mr) * D_;
  const float* brow = W_in + (size_t)(n0 + mr) * D_;   // W_in[n][k]
  for (int k = 0; k < D_; k += 4) {
    f32x2 a, b;
    a.x = arow[k + 2 * kh];
    a.y = arow[k + 2 * kh + 1];
    b.x = brow[k + 2 * kh];
    b.y = brow[k + 2 * kh + 1];
    c = __builtin_amdgcn_wmma_f32_16x16x4_f32(false, a, false, b,
                                              (short)0, c, false, false);
  }
  const int col = lane & 15;
  for (int r = 0; r < 8; ++r) {
    int row = r + 8 * kh;
    target[(size_t)(m0 + row) * D_ + n0 + col] = tanhf(c[r]);
  }
}

// ---------------------------------------------------------------------------
// Kernel 2: fused flash-style pass over context. Each block handles one
// (b, segment) pair: TDM-load 128x256 f32 tile into LDS, compute 128 scores,
// local softmax stats, and the partial weighted context accumulation.
// Context is read from HBM exactly once.
// ---------------------------------------------------------------------------
__global__ __launch_bounds__(256) void sda_attn_partial_kernel(
    const float* __restrict__ context, const float* __restrict__ target,
    float* __restrict__ scores, float* __restrict__ m_p,
    float* __restrict__ s_p, float* __restrict__ acc_p) {
  __shared__ float smem[SEG_ROWS * D_ + 2 * SEG_ROWS];
  float* tile = smem;                        // LDS offset 0 (TDM destination)
  float* sc   = smem + SEG_ROWS * D_;        // 128 raw scores
  float* pv   = sc + SEG_ROWS;               // 128 exp(score - m)

  const int b = blockIdx.x;
  const int seg = blockIdx.y;
  const int l0 = seg * SEG_ROWS;
  const int t = threadIdx.x;
  const int lane = t & 31;
  const int wid = t >> 5;

  const float* src = context + ((size_t)b * L_ + l0) * D_;

  // --- Tensor Data Mover: DMA 128x256 f32 tile (128 KB) into LDS -----------
  if (wid == 0) {
    unsigned long long ga = (unsigned long long)(uintptr_t)src;
    u32x4 g0;
    g0.x = 1u;                                   // count=1, user descriptor
    g0.y = 0u;                                   // lds_addr = 0 (tile base)
    g0.z = (unsigned)(ga & 0xFFFFFFFFu);         // global_addr[31:0]
    g0.w = (unsigned)((ga >> 32) & 0x01FFFFFFu)  // global_addr[56:32]
           | (2u << 30);                         // type = 2 (image)
    i32x8 g1;
    g1[0] = 0x00020000;                          // data_size = 2 -> 4 bytes
    g1[1] = (int)((unsigned)D_ << 16);           // tensor_dim0 lo16 = 256
    g1[2] = (int)((unsigned)SEG_ROWS << 16);     // tensor_dim1 lo16 = 128
    g1[3] = (int)((unsigned)D_ << 16);           // tile_dim0 = 256
    g1[4] = SEG_ROWS;                            // tile_dim1 = 128
    g1[5] = D_;                                  // tensor_dim0_stride = 256
    g1[6] = 0;
    g1[7] = 0;
    i32x4 z4 = {0, 0, 0, 0};
    i32x8 z8 = {0, 0, 0, 0, 0, 0, 0, 0};
    __builtin_amdgcn_tensor_load_to_lds(g0, g1, z4, z4, z8, 0);
    __builtin_amdgcn_s_wait_tensorcnt(0);
  }
  __syncthreads();

  // Per-lane slice of the query vector: d = lane*8 .. lane*8+7
  const float4 tq0 = *(const float4*)(target + (size_t)b * D_ + lane * 8);
  const float4 tq1 = *(const float4*)(target + (size_t)b * D_ + lane * 8 + 4);

  // --- Phase A: scores for 128 rows (wave w handles rows w*16..w*16+15) ----
  for (int rr = 0; rr < 16; ++rr) {
    const int r = wid * 16 + rr;
    const float4* row = (const float4*)(tile + r * D_ + lane * 8);
    float4 x0 = row[0], x1 = row[1];
    float v = x0.x * tq0.x + x0.y * tq0.y + x0.z * tq0.z + x0.w * tq0.w +
              x1.x * tq1.x + x1.y * tq1.y + x1.z * tq1.z + x1.w * tq1.w;
    for (int off = 16; off > 0; off >>= 1) v += __shfl_xor(v, off, 32);
    if (lane == 0) {
      sc[r] = v;
      scores[(size_t)b * L_ + l0 + r] = v;
    }
  }
  __syncthreads();

  // --- Local softmax stats (cheap, redundant per thread) -------------------
  float m = -INFINITY;
  for (int r = 0; r < SEG_ROWS; ++r) m = fmaxf(m, sc[r]);
  if (t < SEG_ROWS) pv[t] = __expf(sc[t] - m);
  __syncthreads();
  float s = 0.f;
  for (int r = 0; r < SEG_ROWS; ++r) s += pv[r];

  // --- Phase B: partial weighted context, thread t owns column d = t -------
  float acc = 0.f;
  for (int r = 0; r < SEG_ROWS; ++r) acc += pv[r] * tile[r * D_ + t];
  acc_p[((size_t)b * SEG + seg) * D_ + t] = acc;
  if (t == 0) {
    m_p[b * SEG + seg] = m;
    s_p[b * SEG + seg] = s;
  }
}

// ---------------------------------------------------------------------------
// Kernel 3: merge the 32 segment partials per batch row (log-sum-exp merge).
// ---------------------------------------------------------------------------
__global__ __launch_bounds__(256) void sda_combine_kernel(
    const float* __restrict__ m_p, const float* __restrict__ s_p,
    const float* __restrict__ acc_p, float* __restrict__ wc,
    float* __restrict__ M_b, float* __restrict__ invS_b) {
  const int b = blockIdx.x;
  const int t = threadIdx.x;   // column d
  float M = -INFINITY;
  for (int s = 0; s < SEG; ++s) M = fmaxf(M, m_p[b * SEG + s]);
  float S = 0.f, a = 0.f;
  for (int s = 0; s < SEG; ++s) {
    const float w = __expf(m_p[b * SEG + s] - M);
    S += s_p[b * SEG + s] * w;
    a += acc_p[((size_t)b * SEG + s) * D_ + t] * w;
  }
  const float inv = 1.f / S;
  wc[(size_t)b * D_ + t] = a * inv;
  if (t == 0) {
    M_b[b] = M;
    invS_b[b] = inv;
  }
}

// ---------------------------------------------------------------------------
// Kernel 4: attn output = exp(score - M_b) / S_b  (1 MB of traffic)
// ---------------------------------------------------------------------------
__global__ __launch_bounds__(256) void sda_attn_norm_kernel(
    const float* __restrict__ scores, const float* __restrict__ M_b,
    const float* __restrict__ invS_b, float* __restrict__ attn) {
  const int i = blockIdx.x * 256 + threadIdx.x;   // 64*4096 total
  const int b = i >> 12;                          // /4096
  attn[i] = __expf(scores[i] - M_b[b]) * invS_b[b];
}

// ---------------------------------------------------------------------------
// Kernel 5: h_tilde = relu(concat([wc, input]) @ W_out^T)  [64,512]x[512,256]^T
// f32 WMMA again, K = 512 split into wc-half and input-half.
// ---------------------------------------------------------------------------
__global__ __launch_bounds__(32) void sda_out_kernel(
    const float* __restrict__ wc, const float* __restrict__ input,
    const float* __restrict__ W_out, float* __restrict__ h_tilde) {
  const int m0 = blockIdx.x * 16;
  const int n0 = blockIdx.y * 16;
  const int lane = threadIdx.x;
  const int mr = lane & 15;
  const int kh = lane >> 4;

  v8f c = {};
  const float* a0 = wc + (size_t)(m0 + mr) * D_;
  const float* a1 = input + (size_t)(m0 + mr) * D_;
  const float* brow = W_out + (size_t)(n0 + mr) * (2 * D_);
  for (int k = 0; k < D_; k += 4) {            // first half: weighted ctx
    f32x2 a, b;
    a.x = a0[k + 2 * kh];
    a.y = a0[k + 2 * kh + 1];
    b.x = brow[k + 2 * kh];
    b.y = brow[k + 2 * kh + 1];
    c = __builtin_amdgcn_wmma_f32_16x16x4_f32(false, a, false, b,
                                              (short)0, c, false, false);
  }
  for (int k = 0; k < D_; k += 4) {            // second half: input
    f32x2 a, b;
    a.x = a1[k + 2 * kh];
    a.y = a1[k + 2 * kh + 1];
    b.x = brow[D_ + k + 2 * kh];
    b.y = brow[D_ + k + 2 * kh + 1];
    c = __builtin_amdgcn_wmma_f32_16x16x4_f32(false, a, false, b,
                                              (short)0, c, false, false);
  }
  const int col = lane & 15;
  for (int r = 0; r < 8; ++r) {
    int row = r + 8 * kh;
    h_tilde[(size_t)(m0 + row) * D_ + n0 + col] = fmaxf(c[r], 0.f);
  }
}

// ---------------------------------------------------------------------------
extern "C" void kernel_launch(void* const* d_in, const int* in_sizes, int n_in,
                              void* d_out, int out_size, void* d_ws,
                              size_t ws_size, hipStream_t stream) {
  const float* input   = (const float*)d_in[0];   // [64,256]
  const float* context = (const float*)d_in[1];   // [64,4096,256]
  const float* W_in    = (const float*)d_in[2];   // [256,256]
  const float* W_out   = (const float*)d_in[3];   // [256,512]
  float* out = (float*)d_out;                     // h_tilde (16384) ++ attn (262144)

  // Workspace layout (floats)
  float* w = (float*)d_ws;
  float* target = w;                         w += B_ * D_;          // 16384
  float* scores = w;                         w += B_ * L_;          // 262144
  float* m_p    = w;                         w += B_ * SEG;         // 2048
  float* s_p    = w;                         w += B_ * SEG;         // 2048
  float* acc_p  = w;                         w += B_ * SEG * D_;    // 524288
  float* wc     = w;                         w += B_ * D_;          // 16384
  float* M_b    = w;                         w += B_;
  float* invS_b = w;                         w += B_;

  float* h_tilde_out = out;
  float* attn_out    = out + B_ * D_;

  // 1) query transform (WMMA f32)
  sda_qt_kernel<<<dim3(B_ / 16, D_ / 16), 32, 0, stream>>>(input, W_in, target);

  // 2) fused scores + partial softmax + partial weighted context (TDM + LDS)
  sda_attn_partial_kernel<<<dim3(B_, SEG), 256, 0, stream>>>(
      context, target, scores, m_p, s_p, acc_p);

  // 3) merge partials
  sda_combine_kernel<<<dim3(B_), 256, 0, stream>>>(m_p, s_p, acc_p, wc, M_b,
                                                   invS_b);

  // 4) normalized attention output
  sda_attn_norm_kernel<<<dim3((B_ * L_) / 256), 256, 0, stream>>>(
      scores, M_b, invS_b, attn_out);

  // 5) output projection + relu (WMMA f32)
  sda_out_kernel<<<dim3(B_ / 16, D_ / 16), 32, 0, stream>>>(wc, input, W_out,
                                                            h_tilde_out);
}